// PlainVoxels_53626961658337
// MI455X (gfx1250) — compile-verified
//
#include <hip/hip_runtime.h>
#include <stdint.h>
#include <math.h>

// VolSDF-style volume render for MI455X (gfx1250, wave32).
// One block (256 threads, 8 waves) per ray; one thread per sample.
// Gather-latency-bound: hash-grid gathers dominate; transmittance scan done
// with wave32 shuffle scan + LDS carry; outputs reduced per block.

#define NUM_RAYS   2048
#define NUM_SAMP   240
#define DT_STEP    0.01f
#define CELL_SZ    0.015f
#define MIN_BETA_C 0.015f
#define TBL_SIZE   20480000     // 40000 * 512
#define EPS_NRM    1e-12f

__global__ __launch_bounds__(256) void volsdf_render_kernel(
    const float* __restrict__ rays_o,
    const float* __restrict__ rays_d,
    const float* __restrict__ rays_d_norm,
    const float* __restrict__ near_p,
    const float* __restrict__ far_p,
    const float* __restrict__ table,
    const float* __restrict__ beta_p,
    float* __restrict__ out)
{
    __shared__ float sbuf[64];   // 8 wave-scan carries, then 8 waves x 8 channels

    const int r    = blockIdx.x;
    const int s    = threadIdx.x;          // sample index (>=NUM_SAMP lanes idle)
    const int lane = threadIdx.x & 31;
    const int wid  = threadIdx.x >> 5;

    // Per-ray uniforms (block-uniform addresses -> scalarized SMEM loads).
    const float ox = rays_o[3*r+0], oy = rays_o[3*r+1], oz = rays_o[3*r+2];
    const float dx = rays_d[3*r+0], dy = rays_d[3*r+1], dz = rays_d[3*r+2];
    const float nr = near_p[r];
    const float fr = far_p[r];
    const float b      = MIN_BETA_C + fabsf(beta_p[0]);
    const float inv_b  = 1.0f / b;        // matches ref: 1.0 / b * (...)

    const float t0 = nr + (float)s * DT_STEP;
    const float t1 = t0 + DT_STEP;
    const float tm = 0.5f * (t0 + t1);
    const bool valid = (s < NUM_SAMP) && (t1 <= fr);

    float tau = 0.0f, alpha = 0.0f;
    float rgb0 = 0.f, rgb1 = 0.f, rgb2 = 0.f;
    float nrx = 0.f, nry = 0.f, nrz = 0.f;

    if (valid) {
        // Skip all table traffic for far-clipped samples (w==0 there anyway).
        const float px = ox + tm * dx;
        const float py = oy + tm * dy;
        const float pz = oz + tm * dz;

        const float ux = px / CELL_SZ;
        const float uy = py / CELL_SZ;
        const float uz = pz / CELL_SZ;
        const float x0x = floorf(ux), x0y = floorf(uy), x0z = floorf(uz);
        const float fx = ux - x0x, fy = uy - x0y, fz = uz - x0z;

        const int bx = (int)x0x, by = (int)x0y, bz = (int)x0z;

        // Spatial hash with int32 wraparound semantics (uint32 mul == XLA int32 mul).
        const uint32_t hx0 = (uint32_t)bx * 73856093u;
        const uint32_t hx1 = hx0 + 73856093u;
        const uint32_t hy0 = (uint32_t)by * 19349663u;
        const uint32_t hy1 = hy0 + 19349663u;
        const uint32_t hz0 = (uint32_t)bz * 83492791u;
        const uint32_t hz1 = hz0 + 83492791u;

        uint32_t idx5[8];
        #pragma unroll
        for (int c = 0; c < 8; ++c) {
            const uint32_t hx = (c & 4) ? hx1 : hx0;
            const uint32_t hy = (c & 2) ? hy1 : hy0;
            const uint32_t hz = (c & 1) ? hz1 : hz0;
            int32_t h = (int32_t)(hx ^ hy ^ hz);
            int32_t m = h % TBL_SIZE;
            m += (m >> 31) & TBL_SIZE;           // python-style non-negative mod
            idx5[c] = (uint32_t)m * 5u;          // row of 5 floats; we use 4
            // gfx1250 global_prefetch_b8: start pulling corner lines while we
            // compute the trilinear weights below.
            __builtin_prefetch((const void*)(table + idx5[c]), 0, 3);
        }

        const float wx[2] = {1.0f - fx, fx};
        const float wy[2] = {1.0f - fy, fy};
        const float wz[2] = {1.0f - fz, fz};

        // Issue all 32 gathers before consuming -> max memory-level parallelism.
        float v[8][4];
        #pragma unroll
        for (int c = 0; c < 8; ++c) {
            const float* p = table + (size_t)idx5[c];
            v[c][0] = p[0];  // sdf
            v[c][1] = p[1];  // r
            v[c][2] = p[2];  // g
            v[c][3] = p[3];  // b   (channel 4 unused by the reference)
        }

        float sdf = 0.f, gx = 0.f, gy = 0.f, gz = 0.f;
        #pragma unroll
        for (int c = 0; c < 8; ++c) {
            const int i = (c >> 2) & 1, j = (c >> 1) & 1, k = c & 1;
            const float wxy = wx[i] * wy[j];
            const float w3  = wxy * wz[k];
            sdf  += w3 * v[c][0];
            rgb0 += w3 * v[c][1];
            rgb1 += w3 * v[c][2];
            rgb2 += w3 * v[c][3];
            // analytic grad of trilinear weight wrt position (chain: du/dp = 1/CELL)
            gx += (i ? 1.f : -1.f) * (wy[j] * wz[k]) * v[c][0];
            gy += (j ? 1.f : -1.f) * (wx[i] * wz[k]) * v[c][0];
            gz += (k ? 1.f : -1.f) * wxy            * v[c][0];
        }
        gx /= CELL_SZ; gy /= CELL_SZ; gz /= CELL_SZ;

        const float gn  = sqrtf(gx*gx + gy*gy + gz*gz);
        const float ivn = 1.0f / fmaxf(gn, EPS_NRM);
        nrx = gx * ivn; nry = gy * ivn; nrz = gz * ivn;

        // Laplace-CDF density (VolSDF)
        const float asdf  = fabsf(sdf);
        const float e     = expm1f(-(asdf / b));
        const float sgn   = (sdf >= 0.f) ? 1.f : -1.f;
        const float sigma = inv_b * (0.5f + 0.5f * sgn * e);
        tau   = sigma * DT_STEP;
        alpha = -expm1f(-tau);
    }

    // ---- block-wide inclusive scan of tau (wave32 shuffle scan + LDS carry) ----
    float inc = tau;
    #pragma unroll
    for (int o = 1; o < 32; o <<= 1) {
        const float ysh = __shfl_up(inc, o, 32);
        if (lane >= o) inc += ysh;
    }
    if (lane == 31) sbuf[wid] = inc;           // wave totals
    __syncthreads();
    float carry = 0.f;
    for (int i = 0; i < wid; ++i) carry += sbuf[i];
    const float excl  = carry + (inc - tau);   // exclusive prefix = cumsum - tau
    const float trans = expf(-excl);
    const float w     = valid ? (trans * alpha) : 0.f;

    // ---- per-sample weighted payload, then 8-channel block reduction ----
    float out8[8];
    out8[0] = w * rgb0;
    out8[1] = w * rgb1;
    out8[2] = w * rgb2;
    out8[3] = w * tm;      // depth numerator (divide by |d| at the end)
    out8[4] = w * nrx;
    out8[5] = w * nry;
    out8[6] = w * nrz;
    out8[7] = w;           // acc

    __syncthreads();       // done reading scan carries; reuse sbuf
    #pragma unroll
    for (int k = 0; k < 8; ++k) {
        float vsum = out8[k];
        #pragma unroll
        for (int o = 16; o > 0; o >>= 1) vsum += __shfl_down(vsum, o, 32);
        if (lane == 0) sbuf[wid * 8 + k] = vsum;
    }
    __syncthreads();

    if (threadIdx.x < 8) {
        float acc = 0.f;
        #pragma unroll
        for (int wg = 0; wg < 8; ++wg) acc += sbuf[wg * 8 + threadIdx.x];
        if (threadIdx.x == 3) acc /= rays_d_norm[r];     // depth /= ||d||
        out[r * 8 + threadIdx.x] = acc;                  // [rgb, depth, nrm, acc]
    }
}

extern "C" void kernel_launch(void* const* d_in, const int* in_sizes, int n_in,
                              void* d_out, int out_size, void* d_ws, size_t ws_size,
                              hipStream_t stream) {
    // setup_inputs() order: rays_o, rays_d, rays_d_norm, near, far, table, beta
    const float* rays_o      = (const float*)d_in[0];
    const float* rays_d      = (const float*)d_in[1];
    const float* rays_d_norm = (const float*)d_in[2];
    const float* near_p      = (const float*)d_in[3];
    const float* far_p       = (const float*)d_in[4];
    const float* table       = (const float*)d_in[5];
    const float* beta_p      = (const float*)d_in[6];
    float* out = (float*)d_out;

    const int n_rays = out_size / 8;   // 2048
    volsdf_render_kernel<<<dim3(n_rays), dim3(256), 0, stream>>>(
        rays_o, rays_d, rays_d_norm, near_p, far_p, table, beta_p, out);
}